// WavePlaneField_76871324664360
// MI455X (gfx1250) — compile-verified
//
#include <hip/hip_runtime.h>

// WavePlaneField for gfx1250 (MI455X).
// Phase 1: transpose planes [6,C,H,W] -> [6,H,W,C] in d_ws (coalesced both
//          sides, global->LDS via gfx1250 async-LDS loads, ASYNCcnt).
// Phase 2: sample. One wave32 per point, lane == channel (FEAT == 32): every
//          bilinear corner fetch is a single 128B coalesced request instead of
//          32 scattered lines. NT stores keep the 201MB transposed plane set
//          resident in the 192MB L2 while streaming the 256MB output.

constexpr int   RESO   = 512;
constexpr int   FEATC  = 32;
constexpr int   HW     = RESO * RESO;
constexpr float INV_B  = 1.0f / 1.6f;     // (p+1.6)/3.2*2-1 == p/1.6
constexpr int   XTILE  = 64;              // x-positions per transpose tile
constexpr int   LROW   = XTILE + 4;       // 68 floats = 272B row: 16B aligned

// Pointer types matching the async-LDS builtin signature:
//   void (int4 addrspace(1)*, int4 addrspace(3)*, imm int, imm int)
typedef int v4i_t __attribute__((__vector_size__(16)));
typedef __attribute__((address_space(1))) v4i_t gv4_t;
typedef __attribute__((address_space(3))) v4i_t lv4_t;

// ---------------------------------------------------------------- transpose
__global__ void __launch_bounds__(256)
wpf_transpose(const float* __restrict__ src, float* __restrict__ dst) {
    __shared__ float tile[FEATC * LROW];            // 8704 B, layout [c][x]

    const int t   = threadIdx.x;
    const int bid = blockIdx.x;                     // p*4096 + y*8 + xb
    const int xb  = bid & 7;
    const int y   = (bid >> 3) & (RESO - 1);
    const int p   = bid >> 12;
    const int x0  = xb * XTILE;

    // Read: each thread pulls two 16B segments (rows c0 and c0+16).
    const int seg = t & 15;                          // 16 segs * 16B = 256B/row
    const int c0  = t >> 4;                          // 0..15

    const float* g0 = src + ((size_t)(p * FEATC + c0     ) * HW
                             + (size_t)y * RESO + x0 + seg * 4);
    const float* g1 = src + ((size_t)(p * FEATC + c0 + 16) * HW
                             + (size_t)y * RESO + x0 + seg * 4);
    float* l0 = &tile[(c0     ) * LROW + seg * 4];   // 16B aligned (LROW=68)
    float* l1 = &tile[(c0 + 16) * LROW + seg * 4];

#if __has_builtin(__builtin_amdgcn_global_load_async_to_lds_b128)
    // Generic LDS addresses carry the LDS byte offset in the low 32 bits
    // (ISA 10.2), so integer round-trips are safe for the AS(3) cast.
    __builtin_amdgcn_global_load_async_to_lds_b128(
        (gv4_t*)(uintptr_t)g0, (lv4_t*)(uintptr_t)l0, 0, 0);
    __builtin_amdgcn_global_load_async_to_lds_b128(
        (gv4_t*)(uintptr_t)g1, (lv4_t*)(uintptr_t)l1, 0, 0);
  #if __has_builtin(__builtin_amdgcn_s_wait_asynccnt)
    __builtin_amdgcn_s_wait_asynccnt(0);
  #else
    asm volatile("s_wait_asynccnt 0x0" ::: "memory");
  #endif
#else
    const float4 v0 = *(const float4*)g0;
    const float4 v1 = *(const float4*)g1;
    *(float4*)l0 = v0;
    *(float4*)l1 = v1;
#endif
    __syncthreads();

    // Write: [c][x] -> global [x][c]; wave reads stride-68 (2-way conflict),
    // wave writes 128B contiguous.
    const size_t rowBase = ((size_t)p * HW + (size_t)y * RESO + x0) * FEATC;
#pragma unroll
    for (int r = 0; r < 8; ++r) {
        int idx = t + r * 256;                       // 0..2047
        int c2  = idx & 31;
        int xl  = idx >> 5;
        dst[rowBase + (size_t)xl * FEATC + c2] = tile[c2 * LROW + xl];
    }
}

// ---------------------------------------------------------------- sampling
struct Samp {
    int   o00, o01, o10, o11;                        // texel index y*RESO+x
    float w00, w01, w10, w11;
};

__device__ __forceinline__ Samp make_samp(float u, float v) {
    const float m = (float)(RESO - 1);
    float x = fminf(fmaxf((u + 1.0f) * 0.5f * m, 0.0f), m);
    float y = fminf(fmaxf((v + 1.0f) * 0.5f * m, 0.0f), m);
    float xf = floorf(x), yf = floorf(y);
    float wx = x - xf,    wy = y - yf;
    int x0 = (int)xf,     y0 = (int)yf;
    int x1 = min(x0 + 1, RESO - 1);
    int y1 = min(y0 + 1, RESO - 1);
    Samp s;
    s.o00 = y0 * RESO + x0;
    s.o01 = y0 * RESO + x1;
    s.o10 = y1 * RESO + x0;
    s.o11 = y1 * RESO + x1;
    float iwx = 1.0f - wx, iwy = 1.0f - wy;
    s.w00 = iwx * iwy;
    s.w01 = wx  * iwy;
    s.w10 = iwx * wy;
    s.w11 = wx  * wy;
    return s;
}

// channels-last: value(c, texel o) at plane_base[o*32 + lane]
__device__ __forceinline__ float bilerp_cl(const float* __restrict__ bp, const Samp& s) {
    float acc = bp[(size_t)(s.o00 << 5)] * s.w00;
    acc = fmaf(bp[(size_t)(s.o01 << 5)], s.w01, acc);
    acc = fmaf(bp[(size_t)(s.o10 << 5)], s.w10, acc);
    acc = fmaf(bp[(size_t)(s.o11 << 5)], s.w11, acc);
    return acc;
}

template <bool CHANNELS_LAST>
__global__ void __launch_bounds__(256)
wpf_sample(const float* __restrict__ pts,
           const float* __restrict__ tim,
           const float* __restrict__ planes,   // CL: [6,H,W,C]; else [6,C,H,W]
           float* __restrict__ out,
           int N) {
    const int lane   = threadIdx.x & 31;
    const int gwave  = (int)((blockIdx.x * blockDim.x + threadIdx.x) >> 5);
    const int nwaves = (int)((gridDim.x * blockDim.x) >> 5);

    // lane's channel offset inside a plane
    const size_t chanOff = CHANNELS_LAST ? (size_t)lane : (size_t)lane * HW;

    for (int n = gwave; n < N; n += nwaves) {
        float px = __builtin_nontemporal_load(pts + (size_t)3 * n + 0);
        float py = __builtin_nontemporal_load(pts + (size_t)3 * n + 1);
        float pz = __builtin_nontemporal_load(pts + (size_t)3 * n + 2);
        float tt = __builtin_nontemporal_load(tim + n);

        float dx = px * INV_B;
        float dy = py * INV_B;
        float dz = pz * INV_B;
        float dt = tt * 2.0f - 1.0f;

        // idx = [[0,1],[0,2],[3,0],[1,2],[3,1],[3,2]]; (u -> W, v -> H)
        Samp s0 = make_samp(dx, dy);
        Samp s1 = make_samp(dx, dz);
        Samp s2 = make_samp(dt, dx);
        Samp s3 = make_samp(dy, dz);
        Samp s4 = make_samp(dt, dy);
        Samp s5 = make_samp(dt, dz);

        float f0, f1, f2, f3, f4, f5;
        if (CHANNELS_LAST) {
            const float* b = planes + chanOff;
            f0 = bilerp_cl(b + (size_t)0 * HW * FEATC, s0);
            f1 = bilerp_cl(b + (size_t)1 * HW * FEATC, s1);
            f2 = bilerp_cl(b + (size_t)2 * HW * FEATC, s2);
            f3 = bilerp_cl(b + (size_t)3 * HW * FEATC, s3);
            f4 = bilerp_cl(b + (size_t)4 * HW * FEATC, s4);
            f5 = bilerp_cl(b + (size_t)5 * HW * FEATC, s5);
        } else {
            const float* b = planes + chanOff;
            auto bl = [&](const float* bp, const Samp& s) {
                float acc = bp[s.o00] * s.w00;
                acc = fmaf(bp[s.o01], s.w01, acc);
                acc = fmaf(bp[s.o10], s.w10, acc);
                acc = fmaf(bp[s.o11], s.w11, acc);
                return acc;
            };
            f0 = bl(b + (size_t)0 * FEATC * HW, s0);
            f1 = bl(b + (size_t)1 * FEATC * HW, s1);
            f2 = bl(b + (size_t)2 * FEATC * HW, s2);
            f3 = bl(b + (size_t)3 * FEATC * HW, s3);
            f4 = bl(b + (size_t)4 * FEATC * HW, s4);
            f5 = bl(b + (size_t)5 * FEATC * HW, s5);
        }

        float space     = f0 * f1 * f3;
        float spacetime = f2 * f4 * f5;

        size_t o = (size_t)n * FEATC + (size_t)lane;
        __builtin_nontemporal_store(space,     out + o);
        __builtin_nontemporal_store(spacetime, out + (size_t)N * FEATC + o);
    }
}

// ---------------------------------------------------------------- launcher
extern "C" void kernel_launch(void* const* d_in, const int* in_sizes, int n_in,
                              void* d_out, int out_size, void* d_ws, size_t ws_size,
                              hipStream_t stream) {
    const float* pts    = (const float*)d_in[0];
    const float* tim    = (const float*)d_in[1];
    const float* planes = (const float*)d_in[2];
    float*       out    = (float*)d_out;

    const int N = in_sizes[1];                       // one time value per point
    const size_t needWs = (size_t)6 * FEATC * HW * sizeof(float);  // 201 MB

    if (ws_size >= needWs) {
        float* planesT = (float*)d_ws;
        // 6 planes * 512 rows * 8 x-blocks = 24576 tiles
        wpf_transpose<<<6 * RESO * (RESO / XTILE), 256, 0, stream>>>(planes, planesT);
        wpf_sample<true><<<8192, 256, 0, stream>>>(pts, tim, planesT, out, N);
    } else {
        wpf_sample<false><<<8192, 256, 0, stream>>>(pts, tim, planes, out, N);
    }
}